// Poolopt_on_Corrmat_58617713655858
// MI455X (gfx1250) — compile-verified
//
#include <hip/hip_runtime.h>
#include <cstdint>
#include <cstring>

// ---------------- problem constants (from reference) ----------------
#define BATCH 32
#define CHN   1024
#define HW    1024          // 32*32 spatial
#define TOPK  256
#define OUTC  258           // 1 max + 1 avg + 256 gathered

typedef float v2f  __attribute__((ext_vector_type(2)));
typedef float v8f  __attribute__((ext_vector_type(8)));
typedef unsigned int v4u __attribute__((ext_vector_type(4)));
typedef int   v4i  __attribute__((ext_vector_type(4)));
typedef int   v8i  __attribute__((ext_vector_type(8)));

// ---- compile-time path probes (visible in hipcc stderr) ----
#if __has_builtin(__builtin_amdgcn_wmma_f32_16x16x4_f32)
#pragma message("PROBE: WMMA f32_16x16x4 builtin PRESENT")
#else
#pragma message("PROBE: WMMA f32_16x16x4 builtin ABSENT -> shuffle fallback")
#endif
#if __has_builtin(__builtin_amdgcn_tensor_load_to_lds) && __has_builtin(__builtin_amdgcn_s_wait_tensorcnt)
#define HAVE_TDM 1
#if defined(__clang_major__) && (__clang_major__ >= 23)
#pragma message("PROBE: TDM builtin PRESENT, using 6-arg form (clang>=23)")
#else
#pragma message("PROBE: TDM builtin PRESENT, using 5-arg form (clang<23)")
#endif
#else
#define HAVE_TDM 0
#pragma message("PROBE: TDM builtin ABSENT -> direct-copy gather")
#endif

// =====================================================================
// Kernel 1: value[b,c] = sum over HW of corr[b,c,:,:]
// One wave (32 lanes) handles 16 channels using V_WMMA_F32_16X16X4_F32:
//   D(16x16) += A(16ch x 4hw) * ones(4x16)  -> every column of D = row sums
// A f32 16x4 layout (ISA 7.12.2): lanes 0-15 = M rows, VGPR0={K0|K2}, VGPR1={K1|K3}
//   => lane L<16 holds (K=0,1) of row L ; lane L+16 holds (K=2,3) of row L
// =====================================================================
__global__ __launch_bounds__(256) void value_sum_kernel(const float* __restrict__ corr,
                                                        float* __restrict__ value) {
    const int wave = blockIdx.x * 8 + (threadIdx.x >> 5);   // 2048 waves total
    const int lane = threadIdx.x & 31;
    const int b  = wave >> 6;          // /64
    const int c0 = (wave & 63) << 4;   // 16 channels per wave
    const float* base = corr + ((size_t)b * CHN + c0) * HW;

#if __has_builtin(__builtin_amdgcn_wmma_f32_16x16x4_f32)
    const int row  = lane & 15;
    const int koff = (lane >> 4) * 2;  // 0 for lanes 0-15, 2 for lanes 16-31
    const float* p = base + (size_t)row * HW + koff;

    v2f ones; ones.x = 1.0f; ones.y = 1.0f;
    v8f acc = {};
#pragma unroll 8
    for (int k0 = 0; k0 < HW; k0 += 4) {
        v2f a = *reinterpret_cast<const v2f*>(p + k0);   // 8B aligned
        // (neg_a, A, neg_b, B, c_mod, C, reuse_a, reuse_b)
        acc = __builtin_amdgcn_wmma_f32_16x16x4_f32(false, a, false, ones,
                                                    (short)0, acc, false, false);
    }
    // D layout: VGPR r -> M=r (lanes 0-15) / M=8+r (lanes 16-31); all N equal
    float* vout = value + b * CHN + c0;
    if (lane == 0) {
#pragma unroll
        for (int r = 0; r < 8; ++r) vout[r] = acc[r];
    } else if (lane == 16) {
#pragma unroll
        for (int r = 0; r < 8; ++r) vout[8 + r] = acc[r];
    }
#else
    // fallback: lane-strided sums + wave reduce
    for (int ch = 0; ch < 16; ++ch) {
        const float* q = base + (size_t)ch * HW;
        float s = 0.0f;
        for (int i = lane; i < HW; i += 32) s += q[i];
        for (int off = 16; off > 0; off >>= 1) s += __shfl_down(s, off, 32);
        if (lane == 0) value[b * CHN + c0 + ch] = s;
    }
#endif
}

// =====================================================================
// Kernel 2: max / mean over channel axis. One thread per (b, hw).
// Coalesced: at fixed c, 256 consecutive threads read 1 KB contiguous.
// corr fits in the 192MB L2, so this second pass mostly hits L2.
// =====================================================================
__global__ __launch_bounds__(256) void max_avg_kernel(const float* __restrict__ corr,
                                                      float* __restrict__ out) {
    const int t  = blockIdx.x * blockDim.x + threadIdx.x;   // 0..32767
    const int b  = t >> 10;
    const int hw = t & 1023;
    const float* p = corr + (size_t)b * CHN * HW + hw;

    float mx = -__builtin_inff();
    float s  = 0.0f;
#pragma unroll 4
    for (int c = 0; c < CHN; ++c) {
        float v = p[(size_t)c * HW];
        __builtin_prefetch(p + (size_t)(c + 16) * HW, 0, 0);  // global_prefetch_b8
        mx = fmaxf(mx, v);
        s += v;
    }
    float* ob = out + (size_t)b * OUTC * HW;
    ob[hw]      = mx;                       // channel 0: max
    ob[HW + hw] = s * (1.0f / (float)CHN);  // channel 1: mean
}

// =====================================================================
// Kernel 3: per-batch bitonic sort of 1024 (value, idx) pairs, descending
// by value (tiebreak: ascending index, matching stable argsort of -value).
// One 1024-thread block per batch; 8 KB LDS.
// =====================================================================
__global__ __launch_bounds__(1024) void topk_sort_kernel(const float* __restrict__ value,
                                                         int* __restrict__ sel) {
    __shared__ float sv[CHN];
    __shared__ int   si[CHN];
    const int b = blockIdx.x;
    const unsigned i = threadIdx.x;

    sv[i] = value[b * CHN + i];
    si[i] = (int)i;

    for (unsigned k = 2; k <= CHN; k <<= 1) {
        for (unsigned j = k >> 1; j > 0; j >>= 1) {
            __syncthreads();
            unsigned ixj = i ^ j;
            if (ixj > i) {
                float vi = sv[i], vj = sv[ixj];
                int   ii = si[i], ij = si[ixj];
                // "precedes" in final order: larger value first, then smaller idx
                bool prec = (vi > vj) || (vi == vj && ii < ij);
                bool up = ((i & k) == 0);
                if (up ? !prec : prec) {
                    sv[i] = vj; sv[ixj] = vi;
                    si[i] = ij; si[ixj] = ii;
                }
            }
        }
    }
    __syncthreads();
    if (i < TOPK) sel[b * TOPK + i] = si[i];
}

// =====================================================================
// Kernel 4: gather selected channel planes (4 KB each) into out[:, 2:, :, :].
// One 256-thread block per (b, k). Where available, stage the 4 KB tile
// through LDS with the Tensor Data Mover (tensor_load_to_lds + TENSORcnt),
// else do a direct float4 copy.
// =====================================================================
__global__ __launch_bounds__(256) void gather_kernel(const float* __restrict__ corr,
                                                     const int* __restrict__ sel,
                                                     float* __restrict__ out) {
    const int b  = blockIdx.x >> 8;
    const int kk = blockIdx.x & 255;
    const int c  = sel[b * TOPK + kk];                       // block-uniform
    const float* src = corr + ((size_t)b * CHN + (size_t)c) * HW;
    float*       dst = out  + ((size_t)b * OUTC + 2 + kk) * HW;

#if HAVE_TDM
    __shared__ float tile[HW];   // only LDS alloc in this kernel -> offset 0

    uint64_t ga = (uint64_t)(uintptr_t)src;
    unsigned galo = (unsigned)__builtin_amdgcn_readfirstlane((int)(ga & 0xffffffffu));
    unsigned gahi = (unsigned)__builtin_amdgcn_readfirstlane((int)(ga >> 32));

    // ---- D# group 0 (128b): count=1 | lds_addr | global_addr[56:0] | type=2
    v4u g0;
    g0.x = 1u;                                    // count=1, user mode
    g0.y = 0u;                                    // lds_addr = 0 (tile @ LDS base)
    g0.z = galo;                                  // global_addr[31:0]
    g0.w = (gahi & 0x01ffffffu) | 0x80000000u;    // global_addr[56:32] | type=2<<30

    // ---- D# group 1 (256b): data_size=4B, tensor 1024x1, tile 1024x1, stride 1024
    v8i g1;
    g1[0] = 0x00020000;           // workgroup_mask=0, data_size=2 (4 bytes)
    g1[1] = (int)(1024u << 16);   // tensor_dim0[15:0]=1024 in bits[63:48]
    g1[2] = (int)(1u << 16);      // tensor_dim0 hi=0 ; tensor_dim1[15:0]=1
    g1[3] = (int)(1024u << 16);   // tensor_dim1 hi=0 ; tile_dim0=1024
    g1[4] = 1;                    // tile_dim1=1 ; tile_dim2=0
    g1[5] = 1024;                 // tensor_dim0_stride[31:0]
    g1[6] = 0;                    // stride hi / tensor_dim1_stride lo
    g1[7] = 0;

    v4i g2 = {0, 0, 0, 0};        // dims 2/3 unused (tile dims zero)
    v4i g3 = {0, 0, 0, 0};

    if (threadIdx.x < 32) {       // first wave issues the DMA (EXEC ignored by TDM)
#if defined(__clang_major__) && (__clang_major__ >= 23)
        v8i g4 = {0, 0, 0, 0, 0, 0, 0, 0};
        __builtin_amdgcn_tensor_load_to_lds(g0, g1, g2, g3, g4, 0);
#else
        __builtin_amdgcn_tensor_load_to_lds(g0, g1, g2, g3, 0);
#endif
        __builtin_amdgcn_s_wait_tensorcnt(0);
    }
    __syncthreads();

    reinterpret_cast<float4*>(dst)[threadIdx.x] =
        reinterpret_cast<const float4*>(tile)[threadIdx.x];
#else
    reinterpret_cast<float4*>(dst)[threadIdx.x] =
        reinterpret_cast<const float4*>(src)[threadIdx.x];
#endif
}

// =====================================================================
// launch
// =====================================================================
extern "C" void kernel_launch(void* const* d_in, const int* in_sizes, int n_in,
                              void* d_out, int out_size, void* d_ws, size_t ws_size,
                              hipStream_t stream) {
    const float* corr = (const float*)d_in[0];
    // d_in[1] = select_indices = arange(256): selection is the identity prefix,
    // so sel_idx == argsort(-value)[:, :256] directly.
    float* out = (float*)d_out;

    float* value = (float*)d_ws;                       // 32*1024 floats
    int*   sel   = (int*)(value + BATCH * CHN);        // 32*256 ints

    // K1: channel sums (WMMA f32 16x16x4). 2048 waves -> 256 blocks x 256 thr.
    value_sum_kernel<<<256, 256, 0, stream>>>(corr, value);
    // K2: max/mean over channels. 32768 threads.
    max_avg_kernel<<<128, 256, 0, stream>>>(corr, out);
    // K3: per-batch bitonic top-256.
    topk_sort_kernel<<<BATCH, 1024, 0, stream>>>(value, sel);
    // K4: gather 32*256 planes.
    gather_kernel<<<BATCH * TOPK, 256, 0, stream>>>(corr, sel, out);
}